// channel_att_80178449482540
// MI455X (gfx1250) — compile-verified
//
#include <hip/hip_runtime.h>
#include <hip/hip_bf16.h>
#include <math.h>

#define INC_C  128
#define BSEG   16
#define HID    32

typedef __attribute__((ext_vector_type(2))) float v2f;
typedef __attribute__((ext_vector_type(4))) float f32x4;
typedef __attribute__((ext_vector_type(8))) float v8f;

// Workspace layout (float units):
//   [0,    2048)  seg_sum   [16][128]
//   [2048, 4096)  seg_max   [16][128]  (float bits, init 0xFF800000 = -inf)
//   [4096, 4112)  counts    [16]       (float)
//   [4112, 6160)  gate      [16][128]
#define WS_SUM  0
#define WS_MAX  2048
#define WS_CNT  4096
#define WS_GATE 4112

__global__ __launch_bounds__(256) void init_ws(float* ws) {
    int i = blockIdx.x * blockDim.x + threadIdx.x;
    if (i < BSEG * INC_C) {
        ws[WS_SUM + i] = 0.0f;
        ((unsigned*)ws)[WS_MAX + i] = 0xFF800000u;   // -inf
    }
    if (i < BSEG) ws[WS_CNT + i] = 0.0f;
}

static __device__ __forceinline__ void atomicMaxF(float* addr, float val) {
    // Monotone-bit trick: signed max for >=0, unsigned min for <0.
    if (val >= 0.0f) atomicMax((int*)addr, __float_as_int(val));
    else             atomicMin((unsigned int*)addr, __float_as_uint(val));
}

static __device__ __forceinline__ f32x4 vmax4(f32x4 a, f32x4 b) {
    f32x4 r;
    r[0] = fmaxf(a[0], b[0]); r[1] = fmaxf(a[1], b[1]);
    r[2] = fmaxf(a[2], b[2]); r[3] = fmaxf(a[3], b[3]);
    return r;
}

static __device__ __forceinline__ void flushSeg(float* ws, int seg, int lane,
                                                f32x4 s, f32x4 mx, int c) {
    int base = seg * INC_C + lane * 4;
#pragma unroll
    for (int j = 0; j < 4; ++j) {
        atomicAdd(&ws[WS_SUM + base + j], s[j]);
        atomicMaxF(&ws[WS_MAX + base + j], mx[j]);
    }
    if (lane == 0) atomicAdd(&ws[WS_CNT + seg], (float)c);
}

// One wave32 per contiguous chunk of rows; lane = 4 channels (512B/row coalesced).
__global__ __launch_bounds__(256) void seg_pool(const float* __restrict__ feats,
                                                const int* __restrict__ bidx,
                                                float* __restrict__ ws,
                                                int n, int chunk) {
    const int lane = threadIdx.x & 31;
    const int wave = blockIdx.x * (blockDim.x >> 5) + (threadIdx.x >> 5);
    int r0 = wave * chunk;
    if (r0 >= n) return;
    int r1 = min(r0 + chunk, n);

    const f32x4* f4 = (const f32x4*)feats;
    f32x4 s  = {0.0f, 0.0f, 0.0f, 0.0f};
    f32x4 mx = {-__builtin_inff(), -__builtin_inff(), -__builtin_inff(), -__builtin_inff()};
    int cur = bidx[r0];
    int c = 0;

    for (int r = r0; r < r1; ++r) {
        int seg = bidx[r];                      // wave-uniform (same addr for all lanes)
        if (seg != cur) {
            flushSeg(ws, cur, lane, s, mx, c);
            s = (f32x4){0.0f, 0.0f, 0.0f, 0.0f};
            mx = (f32x4){-__builtin_inff(), -__builtin_inff(), -__builtin_inff(), -__builtin_inff()};
            c = 0;
            cur = seg;
        }
        f32x4 v = __builtin_nontemporal_load(&f4[(long)r * 32 + lane]);
        s += v;
        mx = vmax4(mx, v);
        ++c;
    }
    flushSeg(ws, cur, lane, s, mx, c);
}

// Single wave: gate = sigmoid( relu(mean@W1+b1)@W2 + relu(max@W1+b1)@W2 + 2*b2 )
// using v_wmma_f32_16x16x4_f32 for both GEMMs.
__global__ __launch_bounds__(32) void mlp_gate(const float* __restrict__ W1,
                                               const float* __restrict__ b1,
                                               const float* __restrict__ W2,
                                               const float* __restrict__ b2,
                                               float* __restrict__ ws) {
    __shared__ float H[BSEG * HID];

    const int lane = threadIdx.x;        // 0..31
    const int m    = lane & 15;          // A/C row group
    const int kh   = (lane >> 4) * 2;    // K-half select for A/B frags
    const int mh   = (lane >> 4) * 8;    // C/D row offset for hi lanes

    const float*    seg_sum  = ws + WS_SUM;
    const unsigned* seg_maxb = (const unsigned*)(ws + WS_MAX);
    const float*    cnt      = ws + WS_CNT;
    float*          gate     = ws + WS_GATE;

    float cm  = cnt[m];
    float inv = 1.0f / fmaxf(cm, 1.0f);
    bool  has = cm > 0.0f;

    // ---- GEMM1: [16x128] @ W1[128x32], two N-tiles, both pooling paths ----
#pragma unroll
    for (int t = 0; t < 2; ++t) {
        v8f accMean = {};
        v8f accMax  = {};
        int n = (lane & 15) + 16 * t;
        for (int k0 = 0; k0 < INC_C; k0 += 4) {
            int k = k0 + kh;
            v2f am, ax, b;
            am[0] = seg_sum[m * INC_C + k]     * inv;
            am[1] = seg_sum[m * INC_C + k + 1] * inv;
            float x0 = __uint_as_float(seg_maxb[m * INC_C + k]);
            float x1 = __uint_as_float(seg_maxb[m * INC_C + k + 1]);
            ax[0] = has ? x0 : 0.0f;
            ax[1] = has ? x1 : 0.0f;
            b[0] = W1[k * HID + n];
            b[1] = W1[(k + 1) * HID + n];
            accMean = __builtin_amdgcn_wmma_f32_16x16x4_f32(false, am, false, b,
                                                            (short)0, accMean, false, false);
            accMax  = __builtin_amdgcn_wmma_f32_16x16x4_f32(false, ax, false, b,
                                                            (short)0, accMax, false, false);
        }
        float b1n = b1[n];
#pragma unroll
        for (int r = 0; r < 8; ++r) {
            int mm = r + mh;
            H[mm * HID + n] = fmaxf(accMean[r] + b1n, 0.0f) + fmaxf(accMax[r] + b1n, 0.0f);
        }
    }
    __syncthreads();

    // ---- GEMM2: H[16x32] @ W2[32x128], eight N-tiles, fuse bias+sigmoid ----
#pragma unroll
    for (int t = 0; t < 8; ++t) {
        v8f acc = {};
        int n = (lane & 15) + 16 * t;
        for (int k0 = 0; k0 < HID; k0 += 4) {
            int k = k0 + kh;
            v2f a, b;
            a[0] = H[m * HID + k];
            a[1] = H[m * HID + k + 1];
            b[0] = W2[k * INC_C + n];
            b[1] = W2[(k + 1) * INC_C + n];
            acc = __builtin_amdgcn_wmma_f32_16x16x4_f32(false, a, false, b,
                                                        (short)0, acc, false, false);
        }
        float b2n = 2.0f * b2[n];
#pragma unroll
        for (int r = 0; r < 8; ++r) {
            int mm = r + mh;
            float z = acc[r] + b2n;
            gate[mm * INC_C + n] = 1.0f / (1.0f + expf(-z));
        }
    }
}

// out[i][c] = feats[i][c] * gate[bidx[i]][c], f32x4 grid-stride stream.
__global__ __launch_bounds__(256) void apply_gate(const float* __restrict__ feats,
                                                  const int* __restrict__ bidx,
                                                  const float* __restrict__ gate,
                                                  float* __restrict__ out,
                                                  int total4) {
    const f32x4* f4 = (const f32x4*)feats;
    const f32x4* g4 = (const f32x4*)gate;
    f32x4*       o4 = (f32x4*)out;
    int stride = gridDim.x * blockDim.x;
    for (int i = blockIdx.x * blockDim.x + threadIdx.x; i < total4; i += stride) {
        int row = i >> 5;               // 32 f32x4 per row
        int c4  = i & 31;
        int seg = bidx[row];
        f32x4 f = __builtin_nontemporal_load(&f4[i]);
        f32x4 g = g4[seg * 32 + c4];    // 8KB table: stays cached
        __builtin_nontemporal_store(f * g, &o4[i]);
    }
}

extern "C" void kernel_launch(void* const* d_in, const int* in_sizes, int n_in,
                              void* d_out, int out_size, void* d_ws, size_t ws_size,
                              hipStream_t stream) {
    const float* feats = (const float*)d_in[0];
    const int*   bidx  = (const int*)d_in[1];
    const float* W1    = (const float*)d_in[2];
    const float* b1    = (const float*)d_in[3];
    const float* W2    = (const float*)d_in[4];
    const float* b2    = (const float*)d_in[5];
    float* out = (float*)d_out;
    float* ws  = (float*)d_ws;
    int n = in_sizes[0] / INC_C;

    init_ws<<<(BSEG * INC_C + 255) / 256, 256, 0, stream>>>(ws);

    int redBlocks  = 512;                       // 8 waves/block -> 4096 waves
    int totalWaves = redBlocks * 8;
    int chunk      = (n + totalWaves - 1) / totalWaves;
    seg_pool<<<redBlocks, 256, 0, stream>>>(feats, bidx, ws, n, chunk);

    mlp_gate<<<1, 32, 0, stream>>>(W1, b1, W2, b2, ws);

    int total4 = n * (INC_C / 4);
    apply_gate<<<16384, 256, 0, stream>>>(feats, bidx, ws + WS_GATE, out, total4);
}